// KeywordsLoss_7035156431386
// MI455X (gfx1250) — compile-verified
//
#include <hip/hip_runtime.h>
#include <hip/hip_bf16.h>
#include <math.h>

// ---------------------------------------------------------------------------
// KeywordsLoss for MI455X (gfx1250, wave32).
// Phase 1: column-mean of logits via V_WMMA_F32_16X16X4_F32 with a parity-
//          selector A matrix: one global_load_b128 per lane feeds TWO WMMAs
//          in place (no cross-register packing moves).
// Phase 2: per-batch max / logsumexp / sum over m (L2-resident).
// Phase 3: closed-form KL(target || logp) using keyword set algebra.
// ---------------------------------------------------------------------------

typedef float v2f  __attribute__((ext_vector_type(2)));
typedef float v4f  __attribute__((ext_vector_type(4)));
typedef float v8f  __attribute__((ext_vector_type(8)));
// V=50257 is odd -> rows are only dword-aligned; allow 4-byte-aligned b128.
typedef v4f v4f_u __attribute__((aligned(4)));

#define KB_B   16
#define KB_S2  256
#define KB_V   50257
#define KB_S1  50
#define KB_ALPHA 0.9f

// ---------------------------------------------------------------------------
// Kernel 1: m[b][v] = mean_s logits[b][s][v], with m[b][0] = 0.
//
// One wave per 64-column tile of one batch.  A is a parity selector:
//   a.x (VGPR0 slot) = 1 if M even else 0;  a.y (VGPR1 slot) = 1 if M odd.
// Lane n (<16) loads float4 of row r at cols v0+4n..4n+3; lane n+16 loads the
// same cols of row r+1.  Feeding B = (r.x,r.y):
//   D[M even][n] += lane n VGPR0 + lane n+16 VGPR0 = colsum(v0+4n)   rows r,r+1
//   D[M odd ][n] += lane n VGPR1 + lane n+16 VGPR1 = colsum(v0+4n+1)
// and B = (r.z,r.w) covers cols 4n+2 / 4n+3.  This holds for ANY
// K<->(VGPR,lane-half) permutation since A and B share the same mapping.
// Result extraction uses the documented D layout: c[0]=M0 (even), c[1]=M1.
// ---------------------------------------------------------------------------
__global__ __launch_bounds__(256)
void kb_colmean_wmma(const float* __restrict__ logits,
                     float* __restrict__ m_out,
                     int ntiles, int nwaves) {
    const int waveId = blockIdx.x * 8 + (threadIdx.x >> 5);
    if (waveId >= nwaves) return;               // wave-uniform: EXEC stays full

    const int b    = waveId / ntiles;
    const int tile = waveId % ntiles;
    const int v0   = tile * 64;
    const int lane = threadIdx.x & 31;

    // Clamp so every lane's 4-column window stays in-bounds (all lanes active
    // for the WMMAs; overlapping lanes recompute identical column sums).
    int colBase = v0 + ((lane & 15) << 2);
    if (colBase > KB_V - 4) colBase = KB_V - 4;
    const int rsel = lane >> 4;                 // lanes 0-15: row r; 16-31: r+1

    const float* __restrict__ src =
        logits + ((size_t)b * KB_S2 + rsel) * KB_V + colBase;

    // Parity-selector A matrix.
    v2f a;
    a.x = (lane & 1) ? 0.0f : 1.0f;             // selects even M rows of D
    a.y = (lane & 1) ? 1.0f : 0.0f;             // selects odd  M rows of D

    v8f c0 = {0.f,0.f,0.f,0.f,0.f,0.f,0.f,0.f}; // cols 4n, 4n+1 (even/odd M)
    v8f c1 = {0.f,0.f,0.f,0.f,0.f,0.f,0.f,0.f}; // cols 4n+2, 4n+3

    for (int k0 = 0; k0 < KB_S2; k0 += 2) {
        v4f r = *(const v4f_u*)(src + (size_t)k0 * KB_V);
        v2f blo = __builtin_shufflevector(r, r, 0, 1);  // in-place subpair
        v2f bhi = __builtin_shufflevector(r, r, 2, 3);  // in-place subpair
        c0 = __builtin_amdgcn_wmma_f32_16x16x4_f32(false, a, false, blo,
                                                   (short)0, c0, false, false);
        c1 = __builtin_amdgcn_wmma_f32_16x16x4_f32(false, a, false, bhi,
                                                   (short)0, c1, false, false);
    }

    // Lane n (<16): c0[0]=colsum(base), c0[1]=colsum(base+1),
    //               c1[0]=colsum(base+2), c1[1]=colsum(base+3).
    if (lane < 16) {
        float vals[4];
        vals[0] = c0[0]; vals[1] = c0[1]; vals[2] = c1[0]; vals[3] = c1[1];
        float* dst = m_out + (size_t)b * KB_V + colBase;
        #pragma unroll
        for (int j = 0; j < 4; ++j) {
            const int v = colBase + j;
            if (v < KB_V)
                dst[j] = (v == 0) ? 0.0f : vals[j] * (1.0f / (float)KB_S2);
        }
    }
}

// ---------------------------------------------------------------------------
// Kernel 2: per-batch stats over m[b][:]:  max, L = log(sum exp(m - max)),
// and summ = sum m.  One 256-thread block per batch.
// ---------------------------------------------------------------------------
__global__ __launch_bounds__(256)
void kb_batch_stats(const float* __restrict__ m, float* __restrict__ stats) {
    const int b   = blockIdx.x;
    const int tid = threadIdx.x;
    const float* __restrict__ row = m + (size_t)b * KB_V;

    __shared__ float red1[256];
    __shared__ float red2[256];

    // ---- max ----
    float mx = -INFINITY;
    for (int v = tid; v < KB_V; v += 256) mx = fmaxf(mx, row[v]);
    red1[tid] = mx;
    __syncthreads();
    for (int s = 128; s > 0; s >>= 1) {
        if (tid < s) red1[tid] = fmaxf(red1[tid], red1[tid + s]);
        __syncthreads();
    }
    mx = red1[0];
    __syncthreads();

    // ---- sum exp(m - max) and sum m ----
    float se = 0.0f, sm = 0.0f;
    for (int v = tid; v < KB_V; v += 256) {
        float x = row[v];
        se += expf(x - mx);
        sm += x;
    }
    red1[tid] = se;
    red2[tid] = sm;
    __syncthreads();
    for (int s = 128; s > 0; s >>= 1) {
        if (tid < s) {
            red1[tid] += red1[tid + s];
            red2[tid] += red2[tid + s];
        }
        __syncthreads();
    }
    if (tid == 0) {
        stats[b * 4 + 0] = mx;
        stats[b * 4 + 1] = logf(red1[0]);
        stats[b * 4 + 2] = red2[0];
    }
}

// ---------------------------------------------------------------------------
// Kernel 3: closed-form loss.  One wave; lanes 0-15 each handle one batch.
//   K = distinct keyword slots after special->0 remap (slot 0 excluded)
//   Z = (V - n) + n*E,  E = e^alpha
//   loss_b = n*E*alpha/Z - logZ - [S_all + (E-1)*S_K]/Z
//   S_all  = summ - V*(max+L);  S_K = sum_{v in K} m[v] - n*(max+L)
// ---------------------------------------------------------------------------
__global__ __launch_bounds__(32)
void kb_final_loss(const float* __restrict__ m,
                   const float* __restrict__ stats,
                   const int* __restrict__ keywords,
                   float* __restrict__ out) {
    const int lane = threadIdx.x;
    __shared__ float lossArr[KB_B];

    if (lane < KB_B) {
        const int b = lane;
        const float mx = stats[b * 4 + 0];
        const float L  = stats[b * 4 + 1];
        const float sm = stats[b * 4 + 2];
        const float c0 = mx + L;                 // logp[v] = m[v] - c0

        int   kws[KB_S1];
        int   n    = 0;
        float S_Km = 0.0f;
        for (int i = 0; i < KB_S1; ++i) {
            int k = keywords[b * KB_S1 + i];
            if (k == 101 || k == 102 || k == 117 || k == 120 || k == 0) k = 0;
            kws[i] = k;
            if (k == 0) continue;                // slot 0 zeroed after scatter
            bool dup = false;
            for (int j = 0; j < i; ++j) {
                if (kws[j] == k) { dup = true; break; }
            }
            if (dup) continue;
            ++n;
            S_Km += m[(size_t)b * KB_V + k];
        }

        const float E    = expf(KB_ALPHA);
        const float Z    = (float)(KB_V - n) + (float)n * E;
        const float logZ = logf(Z);
        const float S_all = sm - (float)KB_V * c0;
        const float S_K   = S_Km - (float)n * c0;
        const float sum_t_logp = (S_all + (E - 1.0f) * S_K) / Z;
        lossArr[b] = ((float)n * E * KB_ALPHA) / Z - logZ - sum_t_logp;
    }
    __syncthreads();
    if (lane == 0) {
        float t = 0.0f;
        for (int i = 0; i < KB_B; ++i) t += lossArr[i];
        out[0] = t / (float)KB_B;                // batchmean
    }
}

// ---------------------------------------------------------------------------
// Launch
// ---------------------------------------------------------------------------
extern "C" void kernel_launch(void* const* d_in, const int* in_sizes, int n_in,
                              void* d_out, int out_size, void* d_ws, size_t ws_size,
                              hipStream_t stream) {
    const float* logits   = (const float*)d_in[0];
    const int*   keywords = (const int*)d_in[1];
    float*       out      = (float*)d_out;

    float* m     = (float*)d_ws;                           // [B, V] fp32
    float* stats = m + (size_t)KB_B * KB_V;                // [B, 4] fp32

    const int ntiles  = (KB_V + 63) / 64;                  // 786 (64-col tiles)
    const int nwaves  = KB_B * ntiles;                     // 12576 waves
    const int blocks1 = (nwaves + 7) / 8;                  // 8 waves / block

    kb_colmean_wmma<<<blocks1, 256, 0, stream>>>(logits, m, ntiles, nwaves);
    kb_batch_stats<<<KB_B, 256, 0, stream>>>(m, stats);
    kb_final_loss<<<1, 32, 0, stream>>>(m, stats, keywords, out);
}